// HintGenKernelBatched_8057358647762
// MI455X (gfx1250) — compile-verified
//
#include <hip/hip_runtime.h>
#include <stdint.h>

typedef long long i64;
typedef unsigned long long u64;
typedef int v4i __attribute__((ext_vector_type(4)));

#define TPB 256
#define TILE 2048   // subset slots staged per LDS tile: 16KB idx + 16KB mask (WGP has 320KB)

#if defined(__gfx1250__) && __has_builtin(__builtin_amdgcn_global_load_async_to_lds_b128)
#define USE_ASYNC_LDS 1
#else
#define USE_ASYNC_LDS 0
#endif

#if USE_ASYNC_LDS
typedef __attribute__((address_space(1))) v4i* gv4p;   // global (AS1) v4i*
typedef __attribute__((address_space(3))) v4i* lv4p;   // LDS (AS3) v4i*

// Flat global pointer -> AS1 pointer: address values are identical on AMDGPU,
// so go through an integer to sidestep const/addrspace C-style-cast pitfalls.
__device__ __forceinline__ gv4p as_global_v4(const void* p) {
  return (gv4p)(unsigned long long)(uintptr_t)p;
}
#endif

__device__ __forceinline__ u64 wave32_xor_reduce(u64 v) {
  // wave32 butterfly; operate on 32-bit halves to avoid 64-bit shuffle overload issues
  #pragma unroll
  for (int w = 16; w >= 1; w >>= 1) {
    unsigned lo = (unsigned)v;
    unsigned hi = (unsigned)(v >> 32);
    lo ^= (unsigned)__shfl_xor((int)lo, w, 32);
    hi ^= (unsigned)__shfl_xor((int)hi, w, 32);
    v = ((u64)hi << 32) | (u64)lo;
  }
  return v;
}

__global__ __launch_bounds__(TPB) void HintGenParityXor_kernel(
    const i64* __restrict__ entries,   // [num_entries, 5]
    const i64* __restrict__ indices,   // [num_hints, max_subset]
    const i64* __restrict__ mask,      // [num_hints, max_subset]
    i64* __restrict__ out,             // [num_hints, 5]
    int max_subset)
{
  __shared__ i64 s_idx[TILE];
  __shared__ i64 s_msk[TILE];
  __shared__ u64 s_acc[5];

  const int hint = blockIdx.x;
  const int tid  = threadIdx.x;

  if (tid < 5) s_acc[tid] = 0ull;

  const i64* idx_row = indices + (size_t)hint * (size_t)max_subset;
  const i64* msk_row = mask    + (size_t)hint * (size_t)max_subset;

  u64 a0 = 0, a1 = 0, a2 = 0, a3 = 0, a4 = 0;

  for (int base = 0; base < max_subset; base += TILE) {
    const int len = (max_subset - base < TILE) ? (max_subset - base) : TILE;

    // ---- stage index + mask tile into LDS ----
#if USE_ASYNC_LDS
    {
      const int bytes   = len * 8;
      const int nchunks = bytes >> 4;          // 16B async chunks
      for (int c = tid; c < nchunks; c += TPB) {
        const int off = c << 4;
        __builtin_amdgcn_global_load_async_to_lds_b128(
            as_global_v4((const char*)(idx_row + base) + off),
            (lv4p)((__attribute__((address_space(3))) char*)s_idx + off),
            0, 0);
        __builtin_amdgcn_global_load_async_to_lds_b128(
            as_global_v4((const char*)(msk_row + base) + off),
            (lv4p)((__attribute__((address_space(3))) char*)s_msk + off),
            0, 0);
      }
      // tail (only if len*8 is not a multiple of 16 — not hit for the given sizes)
      for (int k = (nchunks << 1) + tid; k < len; k += TPB) {
        s_idx[k] = idx_row[base + k];
        s_msk[k] = msk_row[base + k];
      }
      asm volatile("s_wait_asynccnt 0x0" ::: "memory");
    }
#else
    for (int k = tid; k < len; k += TPB) {
      s_idx[k] = idx_row[base + k];
      s_msk[k] = msk_row[base + k];
    }
#endif
    __syncthreads();

    // ---- masked gather from entries (L2-resident, 40MB table) + XOR accumulate ----
    for (int k = tid; k < len; k += TPB) {
      if (s_msk[k] != 0) {
        const i64* e = entries + (size_t)s_idx[k] * 5u;
        a0 ^= (u64)e[0];
        a1 ^= (u64)e[1];
        a2 ^= (u64)e[2];
        a3 ^= (u64)e[3];
        a4 ^= (u64)e[4];
      }
    }
    __syncthreads();
  }

  // ---- wave32 reduction, then cross-wave combine through LDS ----
  a0 = wave32_xor_reduce(a0);
  a1 = wave32_xor_reduce(a1);
  a2 = wave32_xor_reduce(a2);
  a3 = wave32_xor_reduce(a3);
  a4 = wave32_xor_reduce(a4);

  if ((tid & 31) == 0) {
    atomicXor(&s_acc[0], a0);
    atomicXor(&s_acc[1], a1);
    atomicXor(&s_acc[2], a2);
    atomicXor(&s_acc[3], a3);
    atomicXor(&s_acc[4], a4);
  }
  __syncthreads();

  if (tid < 5) {
    out[(size_t)hint * 5u + tid] = (i64)s_acc[tid];
  }
}

extern "C" void kernel_launch(void* const* d_in, const int* in_sizes, int n_in,
                              void* d_out, int out_size, void* d_ws, size_t ws_size,
                              hipStream_t stream) {
  const i64* entries = (const i64*)d_in[0];   // [num_entries, 5] int64
  const i64* indices = (const i64*)d_in[1];   // [num_hints, max_subset] int64
  const i64* mask    = (const i64*)d_in[2];   // [num_hints, max_subset] int64
  i64* out           = (i64*)d_out;           // [num_hints, 5] int64

  const int num_hints  = out_size / 5;
  const int max_subset = (num_hints > 0) ? in_sizes[1] / num_hints : 0;
  if (num_hints <= 0 || max_subset <= 0) return;

  HintGenParityXor_kernel<<<num_hints, TPB, 0, stream>>>(entries, indices, mask, out, max_subset);
}